// StatefulRecurrent_28991029248157
// MI455X (gfx1250) — compile-verified
//
#include <hip/hip_runtime.h>

// StatefulRecurrent (subsampled diagonal complex scan), MI455X / gfx1250.
//
// Reference: s <- A*s + x applied only at odd 0-based t (1024 effective steps);
// even-t output rows are zeros. Memory-bound: 64MB live input + 256MB output
// => ~14us floor at 23.3 TB/s. Strategy:
//   Pass 1: per-chunk zero-init partial states p_c  (C=16 chunks of J=64 steps)
//   Pass 2: carry-in = Horner(A^J, p_0..p_{c-1}); run chunk, emit outputs.
// 16x more wave parallelism; input stays resident in the 192MB L2 between
// passes; outputs use non-temporal stores; inputs staged via gfx1250 async
// global->LDS loads (ASYNCcnt ring, depth 16) when the builtins exist.
//
// NOTE: emit kernel is placed first in the file so the disasm snippet shows it.

namespace {
constexpr int kT = 2048;
constexpr int kB = 8;
constexpr int kD = 2048;
constexpr int kE = kT / 2;            // 1024 effective (updating) steps
constexpr int kJLog2 = 6;
constexpr int kJ = 1 << kJLog2;       // 64 effective steps per chunk
constexpr int kC = kE / kJ;           // 16 chunks
constexpr int kDBlk = 256;            // channels per block (8 waves, wave32)
constexpr int kNDBlk = kD / kDBlk;    // 8 d-blocks
constexpr int kPF = 16;               // async prefetch ring depth (pow2)
constexpr unsigned kBD = (unsigned)kB * kD;   // 16384
constexpr unsigned kInStep = 2u * kBD;        // element stride between effective steps
}  // namespace

typedef __attribute__((ext_vector_type(2))) float f32x2;

#if defined(__HIP_DEVICE_COMPILE__) && defined(__gfx1250__) && \
    __has_builtin(__builtin_amdgcn_global_load_async_to_lds_b32) && \
    __has_builtin(__builtin_amdgcn_s_wait_asynccnt)
#define USE_ASYNC_LDS 1
// Builtin parameter types (per hipcc diagnostic): AS1 int* src, AS3 int* dst.
typedef __attribute__((address_space(1))) int gas_int;
typedef __attribute__((address_space(3))) int las_int;
#else
#define USE_ASYNC_LDS 0
#endif

// ---------------- Pass 2: combine carries and emit all outputs --------------
__global__ __launch_bounds__(kDBlk) void sr_emit_kernel(
    const float* __restrict__ in, const float* __restrict__ Ar_g,
    const float* __restrict__ Ai_g, const f32x2* __restrict__ P,
    f32x2* __restrict__ out2) {
  const int blk = blockIdx.x;
  const int dblk = blk % kNDBlk;
  const int b = (blk / kNDBlk) % kB;
  const int c = blk / (kNDBlk * kB);
  const int d = dblk * kDBlk + threadIdx.x;
  const int tid = threadIdx.x;

  const float Ar = Ar_g[d];
  const float Ai = Ai_g[d];

  // AJ = (Ar + i*Ai)^J by repeated squaring (J = 2^kJLog2)
  float pr = Ar, pi = Ai;
#pragma unroll
  for (int k = 0; k < kJLog2; ++k) {
    const float nr = pr * pr - pi * pi;
    const float ni = 2.0f * pr * pi;
    pr = nr;
    pi = ni;
  }

  // carry-in for chunk c: Horner over preceding chunk partials
  float sr = 0.0f, si = 0.0f;
  for (int k = 0; k < c; ++k) {
    const f32x2 p = P[(k * kB + b) * kD + d];
    const float nr = __builtin_fmaf(pr, sr, __builtin_fmaf(-pi, si, p.x));
    const float ni = __builtin_fmaf(pr, si, __builtin_fmaf(pi, sr, p.y));
    sr = nr;
    si = ni;
  }

  // first effective step of this chunk: t = 2*(c*J) + 1.
  // out2 is indexed in float2 units: (t*B + b)*D + d == same formula as input.
  unsigned gidx = (unsigned)(2 * (c * kJ) + 1) * kBD + (unsigned)b * kD + d;

  f32x2 zz;
  zz.x = 0.0f;
  zz.y = 0.0f;

#if USE_ASYNC_LDS
  __shared__ float xbuf[kPF][kDBlk];
  // Prime the ring: each wave issues kPF async loads (tracked by ASYNCcnt).
#pragma unroll
  for (int jj = 0; jj < kPF; ++jj) {
    __builtin_amdgcn_global_load_async_to_lds_b32(
        (gas_int*)(in + gidx + (unsigned)jj * kInStep),
        (las_int*)&xbuf[jj][tid], 0, 0);
  }
  for (int j = 0; j < kJ; ++j) {
    // async loads complete in order: asynccnt <= kPF-1 => slot j is ready,
    // and only this wave's own lanes wrote this wave's LDS slots (no barrier).
    __builtin_amdgcn_s_wait_asynccnt(kPF - 1);
    const float x = xbuf[j & (kPF - 1)][tid];
    const float nr = __builtin_fmaf(sr, Ar, __builtin_fmaf(-si, Ai, x));
    const float ni = __builtin_fmaf(sr, Ai, si * Ar);
    sr = nr;
    si = ni;
    f32x2 v;
    v.x = sr;
    v.y = si;
    __builtin_nontemporal_store(zz, &out2[gidx - kBD]);  // zero row, even t
    __builtin_nontemporal_store(v, &out2[gidx]);         // data row, odd t
    const int jn = j + kPF;
    if (jn < kJ) {
      __builtin_amdgcn_global_load_async_to_lds_b32(
          (gas_int*)(in + gidx + (unsigned)kPF * kInStep),
          (las_int*)&xbuf[jn & (kPF - 1)][tid], 0, 0);
    }
    gidx += kInStep;
  }
#else
#pragma unroll 4
  for (int j = 0; j < kJ; ++j) {
    const float x = in[gidx];
    __builtin_prefetch(in + gidx + 8u * kInStep, 0, 0);
    const float nr = __builtin_fmaf(sr, Ar, __builtin_fmaf(-si, Ai, x));
    const float ni = __builtin_fmaf(sr, Ai, si * Ar);
    sr = nr;
    si = ni;
    f32x2 v;
    v.x = sr;
    v.y = si;
    __builtin_nontemporal_store(zz, &out2[gidx - kBD]);
    __builtin_nontemporal_store(v, &out2[gidx]);
    gidx += kInStep;
  }
#endif
}

// ---------------- Pass 1: per-chunk partial states (zero initial carry) -----
__global__ __launch_bounds__(kDBlk) void sr_partials_kernel(
    const float* __restrict__ in, const float* __restrict__ Ar_g,
    const float* __restrict__ Ai_g, f32x2* __restrict__ P) {
  const int blk = blockIdx.x;
  const int dblk = blk % kNDBlk;
  const int b = (blk / kNDBlk) % kB;
  const int c = blk / (kNDBlk * kB);
  const int d = dblk * kDBlk + threadIdx.x;

  const float Ar = Ar_g[d];
  const float Ai = Ai_g[d];

  float sr = 0.0f, si = 0.0f;
  // first effective step of this chunk: t = 2*(c*J) + 1
  unsigned gidx = (unsigned)(2 * (c * kJ) + 1) * kBD + (unsigned)b * kD + d;

#pragma unroll 4
  for (int j = 0; j < kJ; ++j) {
    const float x = in[gidx];
    __builtin_prefetch(in + gidx + 8u * kInStep, 0, 0);
    const float nr = __builtin_fmaf(sr, Ar, __builtin_fmaf(-si, Ai, x));
    const float ni = __builtin_fmaf(sr, Ai, si * Ar);
    sr = nr;
    si = ni;
    gidx += kInStep;
  }
  f32x2 p;
  p.x = sr;
  p.y = si;
  P[(c * kB + b) * kD + d] = p;
}

// ---------------- Fallback: monolithic scan (if workspace too small) --------
__global__ __launch_bounds__(kDBlk) void sr_mono_kernel(
    const float* __restrict__ in, const float* __restrict__ Ar_g,
    const float* __restrict__ Ai_g, f32x2* __restrict__ out2) {
  const int dblk = blockIdx.x % kNDBlk;
  const int b = blockIdx.x / kNDBlk;
  const int d = dblk * kDBlk + threadIdx.x;
  const float Ar = Ar_g[d];
  const float Ai = Ai_g[d];
  float sr = 0.0f, si = 0.0f;
  unsigned gidx = kBD + (unsigned)b * kD + d;  // t = 1
  f32x2 zz;
  zz.x = 0.0f;
  zz.y = 0.0f;
#pragma unroll 4
  for (int j = 0; j < kE; ++j) {
    const float x = in[gidx];
    __builtin_prefetch(in + gidx + 8u * kInStep, 0, 0);
    const float nr = __builtin_fmaf(sr, Ar, __builtin_fmaf(-si, Ai, x));
    const float ni = __builtin_fmaf(sr, Ai, si * Ar);
    sr = nr;
    si = ni;
    f32x2 v;
    v.x = sr;
    v.y = si;
    __builtin_nontemporal_store(zz, &out2[gidx - kBD]);
    __builtin_nontemporal_store(v, &out2[gidx]);
    gidx += kInStep;
  }
}

extern "C" void kernel_launch(void* const* d_in, const int* in_sizes, int n_in,
                              void* d_out, int out_size, void* d_ws,
                              size_t ws_size, hipStream_t stream) {
  (void)in_sizes;
  (void)n_in;
  (void)out_size;
  const float* in = (const float*)d_in[0];
  const float* Ar = (const float*)d_in[1];
  const float* Ai = (const float*)d_in[2];
  f32x2* out2 = (f32x2*)d_out;

  const size_t need = (size_t)kC * kB * kD * sizeof(f32x2);  // 2 MB partials
  if (ws_size >= need) {
    f32x2* P = (f32x2*)d_ws;
    dim3 grid(kC * kB * kNDBlk);  // 1024 blocks x 256 thr = 8192 waves
    sr_partials_kernel<<<grid, kDBlk, 0, stream>>>(in, Ar, Ai, P);
    sr_emit_kernel<<<grid, kDBlk, 0, stream>>>(in, Ar, Ai, P, out2);
  } else {
    dim3 grid(kB * kNDBlk);  // 64 blocks, one thread per channel
    sr_mono_kernel<<<grid, kDBlk, 0, stream>>>(in, Ar, Ai, out2);
  }
}